// YoloLossV4_16733192585448
// MI455X (gfx1250) — compile-verified
//
#include <hip/hip_runtime.h>
#include <math.h>

#define NCi 80
#define GRIDB 64

typedef __attribute__((ext_vector_type(2))) float v2f;
typedef __attribute__((ext_vector_type(8))) float v8f;

__device__ __forceinline__ float bce_logits(float x, float t) {
    return fmaxf(x, 0.0f) - x * t + log1pf(expf(-fabsf(x)));
}
__device__ __forceinline__ float sigmoidf_(float x) { return 1.0f / (1.0f + expf(-x)); }

// ---------------------------------------------------------------------------
// Kernel A: target assignment per scale. grid = 3 (one block per scale),
// block = 512 threads (one per flat target). Computes per-scale
// lbox_i, lcls_i and the lobj correction  corr_i = -sum_{unique positives} p_obj.
// ---------------------------------------------------------------------------
__global__ void yolo_targets_kernel(const float* __restrict__ p0,
                                    const float* __restrict__ p1,
                                    const float* __restrict__ p2,
                                    const float* __restrict__ targets,
                                    const float* __restrict__ anchors,
                                    const float* __restrict__ image_size,
                                    int NT, float* __restrict__ scaleRes)
{
    const int scale = blockIdx.x;
    const int dims[3] = {80, 40, 20};
    const float* pp = (scale == 0) ? p0 : ((scale == 1) ? p1 : p2);
    const int H = dims[scale], W = dims[scale];
    const int hw = H * W;
    const float strideH = image_size[0] / (float)H;
    const float strideW = image_size[1] / (float)W;
    const float gainX = 1.0f / strideW;   // gain[2] == gain[4]
    const float gainY = 1.0f / strideH;   // gain[3] == gain[5]
    float aw[3], ah[3];
    for (int a = 0; a < 3; a++) {
        aw[a] = anchors[scale * 6 + a * 2 + 0] / strideW;
        ah[a] = anchors[scale * 6 + a * 2 + 1] / strideH;
    }

    __shared__ int    skey[512];
    __shared__ float4 sred[512];
    const int tid = threadIdx.x;

    float lb = 0.0f, lc = 0.0f, nbv = 0.0f, corr = 0.0f;
    int   mykey = -1;
    bool  mask = false;
    int   b = 0, abest = 0, gi = 0, gj = 0;
    float t1 = 0, t2 = 0, t3 = 0, t4 = 0, t5 = 0;

    if (tid < NT) {
        const float* tr = targets + (size_t)tid * 6;
        float t0 = tr[0];
        t1 = tr[1];
        t2 = tr[2] * gainX;
        t3 = tr[3] * gainY;
        t4 = tr[4] * gainX;
        t5 = tr[5] * gainY;
        float best = -1e30f;
        abest = 0;
        for (int a = 0; a < 3; a++) {   // argmax keeps FIRST max (strict >), matches jnp
            float inter = fminf(aw[a], t4) * fminf(ah[a], t5);
            float uni   = aw[a] * ah[a] + t4 * t5 - inter;
            float iou   = inter / uni;
            if (iou > best) { best = iou; abest = a; }
        }
        mask = best > 0.6f;
        if (mask) {
            b  = (int)t0;
            gi = (int)t2;
            gj = (int)t3;
            mykey = ((b * 3 + abest) * H + gj) * W + gi;
        }
    }
    skey[tid] = mykey;
    __syncthreads();

    bool owner = mask;                    // de-dup for the tobj scatter-max
    if (mask) {
        for (int k = 0; k < tid; k++)
            if (skey[k] == mykey) { owner = false; break; }
    }

    if (mask) {
        size_t baseOff = (((size_t)b * 255 + (size_t)abest * 85) * hw) + (size_t)gj * W + gi;
        float ps0 = pp[baseOff + 0 * (size_t)hw];
        float ps1 = pp[baseOff + 1 * (size_t)hw];
        float ps2 = pp[baseOff + 2 * (size_t)hw];
        float ps3 = pp[baseOff + 3 * (size_t)hw];
        float ps4 = pp[baseOff + 4 * (size_t)hw];
        // GIoU
        float pbx = sigmoidf_(ps0);
        float pby = sigmoidf_(ps1);
        float pbw = fminf(expf(ps2), 1000.0f) * aw[abest];
        float pbh = fminf(expf(ps3), 1000.0f) * ah[abest];
        float tbx = t2 - floorf(t2);
        float tby = t3 - floorf(t3);
        float tbw = t4, tbh = t5;
        float px1 = pbx - pbw * 0.5f, py1 = pby - pbh * 0.5f;
        float px2 = pbx + pbw * 0.5f, py2 = pby + pbh * 0.5f;
        float tx1 = tbx - tbw * 0.5f, ty1 = tby - tbh * 0.5f;
        float tx2 = tbx + tbw * 0.5f, ty2 = tby + tbh * 0.5f;
        float iw = fmaxf(fminf(px2, tx2) - fmaxf(px1, tx1), 0.0f);
        float ih = fmaxf(fminf(py2, ty2) - fmaxf(py1, ty1), 0.0f);
        float inter = iw * ih;
        float uni   = pbw * pbh + tbw * tbh - inter + 1e-7f;
        float iou   = inter / uni;
        float cw  = fmaxf(px2, tx2) - fminf(px1, tx1);
        float chh = fmaxf(py2, ty2) - fminf(py1, ty1);
        float c_area = cw * chh + 1e-7f;
        float giou = iou - (c_area - uni) / c_area;
        lb = 1.0f - giou;
        // classification BCE vs one-hot(c-1)
        int cm1 = (int)t1 - 1;
        for (int cc = 0; cc < NCi; cc++) {
            float x = pp[baseOff + (size_t)(5 + cc) * hw];
            lc += bce_logits(x, (cc == cm1) ? 1.0f : 0.0f);
        }
        nbv = 1.0f;
        if (owner) corr = -ps4;           // BCE(x,1)-BCE(x,0) == -x
    }

    sred[tid] = make_float4(lb, lc, nbv, corr);
    __syncthreads();
    for (int s = 256; s > 0; s >>= 1) {   // deterministic tree reduction
        if (tid < s) {
            float4 x = sred[tid], y = sred[tid + s];
            sred[tid] = make_float4(x.x + y.x, x.y + y.y, x.z + y.z, x.w + y.w);
        }
        __syncthreads();
    }
    if (tid == 0) {
        float4 r = sred[0];
        float nb = fmaxf(r.z, 1.0f);
        scaleRes[scale * 4 + 0] = r.x / nb;                   // lbox_i
        scaleRes[scale * 4 + 1] = r.y / (nb * (float)NCi);    // lcls_i
        scaleRes[scale * 4 + 2] = r.w;                        // corr_i
        scaleRes[scale * 4 + 3] = 0.0f;
    }
}

// ---------------------------------------------------------------------------
// Kernel B: streaming sum of BCE(p_obj, 0) over all B*3*H*W obj positions.
// Accumulation runs on the matrix pipe: D = ones(16x4) x B(4x16) + C keeps
// column sums of each 64-value batch in every row of C (f32, RNE). Row M=0
// (accumulator VGPR 0, lanes 0-15) carries the wave's partial sum.
// EXEC stays all-ones inside the loop (uniform bounds, clamped loads, selects).
// ---------------------------------------------------------------------------
__global__ void obj_bce_kernel(const float* __restrict__ p, int hw, int N,
                               float* __restrict__ partials)
{
    const int tid = threadIdx.x;
    v8f c = {};
    v2f av = {1.0f, 1.0f};
    for (int base = blockIdx.x * 512; base < N; base += GRIDB * 512) {
        int i0 = base + tid;
        int i1 = base + 256 + tid;
        int e0 = (i0 < N) ? i0 : (N - 1);
        int e1 = (i1 < N) ? i1 : (N - 1);
        int ba0 = e0 / hw, yx0 = e0 % hw;
        int ba1 = e1 / hw, yx1 = e1 % hw;
        int b0 = ba0 / 3, a0 = ba0 % 3;
        int b1 = ba1 / 3, a1 = ba1 % 3;
        float x0 = p[((size_t)b0 * 255 + (size_t)a0 * 85 + 4) * (size_t)hw + yx0];
        float x1 = p[((size_t)b1 * 255 + (size_t)a1 * 85 + 4) * (size_t)hw + yx1];
        float bce0 = fmaxf(x0, 0.0f) + log1pf(expf(-fabsf(x0)));
        float bce1 = fmaxf(x1, 0.0f) + log1pf(expf(-fabsf(x1)));
        bce0 = (i0 < N) ? bce0 : 0.0f;
        bce1 = (i1 < N) ? bce1 : 0.0f;
        v2f bv = {bce0, bce1};
        // 8 args: (neg_a, A, neg_b, B, c_mod, C, reuse_a, reuse_b)
        c = __builtin_amdgcn_wmma_f32_16x16x4_f32(false, av, false, bv, (short)0, c,
                                                  false, false);
    }
    __shared__ float sdata[256];
    int lane = tid & 31;
    sdata[tid] = (lane < 16) ? c[0] : 0.0f;   // row M=0 colsums only
    __syncthreads();
    for (int s = 128; s > 0; s >>= 1) {
        if (tid < s) sdata[tid] += sdata[tid + s];
        __syncthreads();
    }
    if (tid == 0) partials[blockIdx.x] = sdata[0];
}

// ---------------------------------------------------------------------------
// Finalize: fixed-order sums, weights, 4 scalar outputs.
// ---------------------------------------------------------------------------
__global__ void finalize_kernel(const float* __restrict__ scaleRes,
                                const float* __restrict__ partials,
                                int Bsz, float* __restrict__ out)
{
    if (threadIdx.x == 0 && blockIdx.x == 0) {
        const int dims[3] = {80, 40, 20};
        float lbox = 0.0f, lobj = 0.0f, lcls = 0.0f;
        for (int s = 0; s < 3; s++) {
            float sum = 0.0f;
            for (int k = 0; k < GRIDB; k++) sum += partials[s * GRIDB + k];
            float N = (float)(Bsz * 3 * dims[s] * dims[s]);
            lobj += (sum + scaleRes[s * 4 + 2]) / N;
            lbox += scaleRes[s * 4 + 0];
            lcls += scaleRes[s * 4 + 1];
        }
        lbox *= 3.54f; lobj *= 64.3f; lcls *= 37.4f;
        out[0] = lbox + lobj + lcls;
        out[1] = lbox;
        out[2] = lobj;
        out[3] = lcls;
    }
}

extern "C" void kernel_launch(void* const* d_in, const int* in_sizes, int n_in,
                              void* d_out, int out_size, void* d_ws, size_t ws_size,
                              hipStream_t stream)
{
    const float* p0         = (const float*)d_in[0];
    const float* p1         = (const float*)d_in[1];
    const float* p2         = (const float*)d_in[2];
    const float* targets    = (const float*)d_in[3];
    const float* anchors    = (const float*)d_in[4];
    const float* image_size = (const float*)d_in[5];
    float* out = (float*)d_out;

    float* scaleRes = (float*)d_ws;        // 16 floats
    float* partials = scaleRes + 16;       // 3*GRIDB floats

    int NT = in_sizes[3] / 6;              // flat target rows (512)
    if (NT > 512) NT = 512;
    int Bsz = in_sizes[0] / (255 * 80 * 80);   // batch (16)

    yolo_targets_kernel<<<3, 512, 0, stream>>>(p0, p1, p2, targets, anchors,
                                               image_size, NT, scaleRes);

    const int dims[3] = {80, 40, 20};
    const float* pps[3] = {p0, p1, p2};
    for (int s = 0; s < 3; s++) {
        int hw = dims[s] * dims[s];
        int N  = Bsz * 3 * hw;
        obj_bce_kernel<<<GRIDB, 256, 0, stream>>>(pps[s], hw, N, partials + s * GRIDB);
    }
    finalize_kernel<<<1, 32, 0, stream>>>(scaleRes, partials, Bsz, out);
}